// PytorchModel_12369505812695
// MI455X (gfx1250) — compile-verified
//
#include <hip/hip_runtime.h>
#include <hip/hip_bf16.h>

// DeltaNet single recurrent step, algebraically reduced:
//   out[d] = decay*qts[d] + (q·k)*beta*(v[d] - decay*kts[d])
// where [qts; kts] = [q; k] @ state   (two matvecs fused into f32 WMMA).
//
// Roofline: one pass over state = 201 MB @ 23.3 TB/s ~ 8.6 us floor; 0.2 GFLOP
// -> memory-bound. The WMMA B lane layout wants 4B/lane reads (128B/request;
// 182 G req/s at peak BW). We instead stream state global->LDS with async
// per-lane b128 copies (512B/request, 45 G req/s) and feed WMMA from LDS,
// double-buffered on ASYNCcnt. LDS row stride padded to 136 dwords so the two
// half-waves of each B read hit disjoint bank groups (conflict-free).

typedef float v2f __attribute__((ext_vector_type(2)));
typedef float v8f __attribute__((ext_vector_type(8)));

#if __has_builtin(__builtin_amdgcn_global_load_async_to_lds_b128)
#define HAVE_ASYNC_LDS 1
#else
#define HAVE_ASYNC_LDS 0
#endif

#define B_DIM 64
#define H_DIM 48
#define BH_TOTAL (B_DIM * H_DIM)
#define DK 128
#define DV 128
#define WAVES_PER_BLOCK 4
#define QK_SLICE 264            // floats: 128 q + 128 k + 2 zeros + pad
#define ROWF 136                // padded LDS row stride (dwords)
#define ROWS_PER_STAGE 8
#define STAGE_F (ROWS_PER_STAGE * ROWF)      // 1088 floats per stage buffer
#define WAVE_LDS (QK_SLICE + 2 * STAGE_F)    // 2440 floats per wave
#define NSTAGES (DK / ROWS_PER_STAGE)        // 16

template <int N>
__device__ __forceinline__ void wait_async() {
#if HAVE_ASYNC_LDS
#if __has_builtin(__builtin_amdgcn_s_wait_asynccnt)
    __builtin_amdgcn_s_wait_asynccnt(N);
#else
    asm volatile("s_wait_asynccnt %0" ::"i"(N) : "memory");
#endif
    asm volatile("" ::: "memory");  // keep LDS reads after the wait
#endif
}

#if HAVE_ASYNC_LDS
// Exact parameter types per the compiler diagnostic: pointer to
// int __attribute__((vector_size(16))), global (AS1) src / LDS (AS3) dst.
typedef int v4i_vs __attribute__((vector_size(16)));
typedef __attribute__((address_space(1))) v4i_vs* gbl_v4i_p;
typedef __attribute__((address_space(3))) v4i_vs* lds_v4i_p;

__device__ __forceinline__ void async_row_copy(float* ldsdst, const float* gsrc) {
    __builtin_amdgcn_global_load_async_to_lds_b128(
        (gbl_v4i_p)gsrc, (lds_v4i_p)ldsdst, 0, 0);
}

// One stage = 8 consecutive state rows (4 KB). One async b128 per row:
// 32 lanes x 16 B = 512 B = exactly one row, fully coalesced.
__device__ __forceinline__ void issue_stage(float* buf, const float* gbase, int lane) {
#pragma unroll
    for (int r = 0; r < ROWS_PER_STAGE; ++r)
        async_row_copy(buf + r * ROWF + lane * 4, gbase + r * DV + lane * 4);
}
#endif

__global__ __launch_bounds__(WAVES_PER_BLOCK * 32) void deltanet_step_kernel(
    const float* __restrict__ qg, const float* __restrict__ kg,
    const float* __restrict__ vg, const float* __restrict__ betag,
    const float* __restrict__ gateg, const float* __restrict__ stateg,
    __hip_bfloat16* __restrict__ outg)
{
    __shared__ float lds[WAVES_PER_BLOCK * WAVE_LDS];

    const int wid  = (int)(threadIdx.x >> 5);
    const int lane = (int)(threadIdx.x & 31);
    const int bh   = (int)blockIdx.x * WAVES_PER_BLOCK + wid;  // grid sized exactly

    float* slice = lds + wid * WAVE_LDS;  // [0..127]=q, [128..255]=k, [256..257]=0
    const float* qp = qg + (size_t)bh * DK;
    const float* kp = kg + (size_t)bh * DK;

    // Stage q,k into this wave's LDS slice (float4 per lane, 16B aligned).
    float4 q4 = ((const float4*)qp)[lane];
    float4 k4 = ((const float4*)kp)[lane];
    ((float4*)slice)[lane]        = q4;
    ((float4*)(slice + DK))[lane] = k4;
    if (lane == 0) { slice[256] = 0.0f; slice[257] = 0.0f; }

    // q·k via wave32 xor-shuffle reduction.
    float qk = q4.x * k4.x + q4.y * k4.y + q4.z * k4.z + q4.w * k4.w;
#pragma unroll
    for (int off = 16; off > 0; off >>= 1) qk += __shfl_xor(qk, off, 32);

    const float bet = betag[bh];
    const float dec = expf(gateg[bh]);

    __syncthreads();  // LDS q/k visible before per-lane ds_load of A tiles

    // A-matrix 16x4 f32 layout: lane m=lane&15 -> row M, half-wave hl -> K pair.
    // Row 0 = q, row 1 = k, rows 2..15 = 0 (read the LDS zero slot).
    const int m  = lane & 15;
    const int hl = lane >> 4;
    const float* abase     = (m == 0) ? slice : (m == 1) ? (slice + DK) : (slice + 256);
    const int    a_advance = (m < 2) ? 1 : 0;

    const float* sb = stateg + (size_t)bh * (DK * DV);

    v8f acc[8] = {};  // 8 column tiles of 16 -> DV=128; D rows M=0(qts), M=1(kts)

#if HAVE_ASYNC_LDS
    float* stg0 = slice + QK_SLICE;
    float* stg1 = slice + QK_SLICE + STAGE_F;

    issue_stage(stg0, sb, lane);

    for (int s = 0; s < NSTAGES; ++s) {
        float* buf = (s & 1) ? stg1 : stg0;
        if (s + 1 < NSTAGES) {
            float* nbuf = (s & 1) ? stg0 : stg1;  // consumed in the previous stage (s-1)
            issue_stage(nbuf, sb + (size_t)(s + 1) * ROWS_PER_STAGE * DV, lane);
            wait_async<ROWS_PER_STAGE>();         // stage s complete, s+1 in flight
        } else {
            wait_async<0>();
        }
        const int k0 = s * ROWS_PER_STAGE;
#pragma unroll
        for (int ks = 0; ks < 2; ++ks) {
            const int kubl = 4 * ks + 2 * hl;  // local K base for this half-wave
            v2f a = *(const v2f*)(abase + (a_advance ? (k0 + kubl) : 0));
            const float* bp = buf + kubl * ROWF + m;
#pragma unroll
            for (int t = 0; t < 8; ++t) {
                v2f bmat;
                bmat.x = bp[t * 16];
                bmat.y = bp[t * 16 + ROWF];
                acc[t] = __builtin_amdgcn_wmma_f32_16x16x4_f32(
                    false, a, false, bmat, (short)0, acc[t], false, false);
            }
        }
    }
#else
    // Fallback: feed WMMA B directly from global (coalesced b32).
    for (int step = 0; step < 32; ++step) {
        const int k0  = step * 4;
        const int kub = k0 + 2 * hl;
        v2f a = *(const v2f*)(abase + (a_advance ? kub : 0));
        const float* bp = sb + (size_t)kub * DV + m;
#pragma unroll
        for (int t = 0; t < 8; ++t) {
            v2f bmat;
            bmat.x = bp[t * 16];
            bmat.y = bp[t * 16 + DV];
            acc[t] = __builtin_amdgcn_wmma_f32_16x16x4_f32(
                false, a, false, bmat, (short)0, acc[t], false, false);
        }
    }
#endif

    // Epilogue: qts = acc[t][0] (M=0), kts = acc[t][1] (M=1), valid in lanes 0-15.
    const float scale = qk * bet;
    if (lane < 16) {
        __hip_bfloat16* op = outg + (size_t)bh * DV;
        const float*    vp = vg   + (size_t)bh * DV;
#pragma unroll
        for (int t = 0; t < 8; ++t) {
            float vv  = vp[t * 16 + lane];
            float qts = acc[t][0];
            float kts = acc[t][1];
            float o   = dec * qts + scale * (vv - dec * kts);
            op[t * 16 + lane] = __float2bfloat16(o);
        }
    }
}

extern "C" void kernel_launch(void* const* d_in, const int* in_sizes, int n_in,
                              void* d_out, int out_size, void* d_ws, size_t ws_size,
                              hipStream_t stream) {
    (void)in_sizes; (void)n_in; (void)d_ws; (void)ws_size; (void)out_size;
    const float* q     = (const float*)d_in[0];
    const float* k     = (const float*)d_in[1];
    const float* v     = (const float*)d_in[2];
    const float* beta  = (const float*)d_in[3];
    const float* gate  = (const float*)d_in[4];
    const float* state = (const float*)d_in[5];
    __hip_bfloat16* out = (__hip_bfloat16*)d_out;

    dim3 grid(BH_TOTAL / WAVES_PER_BLOCK);  // 768 blocks, 1 wave per (b,h)
    dim3 block(WAVES_PER_BLOCK * 32);       // 128 threads = 4 wave32
    deltanet_step_kernel<<<grid, block, 0, stream>>>(q, k, v, beta, gate, state, out);
}